// TUPEMultiHeadAttention_15272903705328
// MI455X (gfx1250) — compile-verified
//
#include <hip/hip_runtime.h>
#include <hip/hip_bf16.h>
#include <math.h>
#include <stdint.h>

typedef __attribute__((ext_vector_type(2))) float v2f;
typedef __attribute__((ext_vector_type(8))) float v8f;
typedef __attribute__((ext_vector_type(4))) unsigned int u32x4;
typedef __attribute__((ext_vector_type(8))) unsigned int u32x8;

constexpr int Bc = 2;
constexpr int Sc = 2048;
constexpr int Dc = 1024;
constexpr int Hc = 16;
constexpr int DHc = 64;

static __device__ __forceinline__ v8f wmma_f32(v2f a, v2f b, v8f c) {
  // D(16x16,f32) = A(16x4,f32) * B(4x16,f32) + C
  return __builtin_amdgcn_wmma_f32_16x16x4_f32(false, a, false, b, (short)0, c,
                                               false, false);
}

// ---------------------------------------------------------------------------
// TDM: async 2D tile DMA global -> LDS (2-SGPR-group descriptor form).
// tile_w contiguous elements per row, tile_h rows, row stride in elements,
// optional LDS padding (interval/amount encoded per D# group1 spec).
// Tracked by TENSORcnt; caller must s_wait_tensorcnt before consuming.
// ---------------------------------------------------------------------------
static __device__ __forceinline__ void tdm_load_2d(
    unsigned lds_off, const float* gptr, unsigned tile_w, unsigned tile_h,
    unsigned row_stride_elems, unsigned pad_enable, unsigned pad_interval_code,
    unsigned pad_amount_code) {
  unsigned long long ga = (unsigned long long)(uintptr_t)gptr;
  u32x4 g0;
  g0[0] = 1u;                                             // count=1, user mode
  g0[1] = lds_off;                                        // lds_addr (bytes)
  g0[2] = (unsigned)(ga & 0xffffffffu);                   // global_addr[31:0]
  g0[3] = (unsigned)((ga >> 32) & 0x01ffffffu) | (2u << 30);  // [56:32] | type=2
  u32x8 g1;
  g1[0] = (2u << 16)                                      // data_size = 4B
        | (pad_enable << 20)
        | (pad_interval_code << 22)
        | (pad_amount_code << 25);                        // workgroup_mask=0
  g1[1] = (tile_w & 0xffffu) << 16;                       // tensor_dim0[15:0]
  g1[2] = ((tile_w >> 16) & 0xffffu) | ((tile_h & 0xffffu) << 16); // d0 hi | d1 lo
  g1[3] = ((tile_h >> 16) & 0xffffu) | ((tile_w & 0xffffu) << 16); // d1 hi | tile_dim0
  g1[4] = (tile_h & 0xffffu);                             // tile_dim1, tile_dim2=0
  g1[5] = row_stride_elems;                               // tensor_dim0_stride[31:0]
  g1[6] = 0u;
  g1[7] = 0u;
  asm volatile("tensor_load_to_lds %0, %1" :: "s"(g0), "s"(g1) : "memory");
}

// ---------------------------------------------------------------------------
// Generic fp32 WMMA GEMM:  C[M,N] = A[M,K] * W[K,N]
// block = 128 threads (4 waves), tile 64x64, K-step 16, LDS staged.
// ---------------------------------------------------------------------------
__global__ __launch_bounds__(128) void wmma_gemm_f32(
    const float* __restrict__ A, const float* __restrict__ W,
    float* __restrict__ C, int M, int N, int K) {
  __shared__ float As[64][17];
  __shared__ float Bs[16][68];

  const int tid = threadIdx.x;
  const int wave = tid >> 5;
  const int lane = tid & 31;
  const int half = lane >> 4;
  const int l = lane & 15;
  const int m0 = blockIdx.y * 64;
  const int n0 = blockIdx.x * 64;
  const int wm = (wave >> 1) * 32;
  const int wn = (wave & 1) * 32;

  v8f acc[2][2] = {};

  for (int k0 = 0; k0 < K; k0 += 16) {
    __syncthreads();
    for (int i = 0; i < 2; ++i) {
      int idx = tid * 2 + i;
      int r = idx >> 2, c4 = (idx & 3) * 4;
      float4 v = *(const float4*)(A + (size_t)(m0 + r) * K + k0 + c4);
      As[r][c4 + 0] = v.x; As[r][c4 + 1] = v.y;
      As[r][c4 + 2] = v.z; As[r][c4 + 3] = v.w;
    }
    for (int i = 0; i < 2; ++i) {
      int idx = tid * 2 + i;
      int r = idx >> 4, c4 = (idx & 15) * 4;
      float4 v = *(const float4*)(W + (size_t)(k0 + r) * N + n0 + c4);
      Bs[r][c4 + 0] = v.x; Bs[r][c4 + 1] = v.y;
      Bs[r][c4 + 2] = v.z; Bs[r][c4 + 3] = v.w;
    }
    __syncthreads();

    for (int kk = 0; kk < 16; kk += 4) {
      v2f a[2], b[2];
      for (int mi = 0; mi < 2; ++mi) {
        int row = wm + mi * 16 + l;
        a[mi].x = As[row][kk + 2 * half + 0];
        a[mi].y = As[row][kk + 2 * half + 1];
      }
      for (int ni = 0; ni < 2; ++ni) {
        int col = wn + ni * 16 + l;
        b[ni].x = Bs[kk + 2 * half + 0][col];
        b[ni].y = Bs[kk + 2 * half + 1][col];
      }
      for (int mi = 0; mi < 2; ++mi)
        for (int ni = 0; ni < 2; ++ni)
          acc[mi][ni] = wmma_f32(a[mi], b[ni], acc[mi][ni]);
    }
  }

  for (int mi = 0; mi < 2; ++mi)
    for (int ni = 0; ni < 2; ++ni)
      for (int j = 0; j < 8; ++j) {
        int row = m0 + wm + mi * 16 + j + 8 * half;
        int col = n0 + wn + ni * 16 + l;
        C[(size_t)row * N + col] = acc[mi][ni][j];
      }
}

// ---------------------------------------------------------------------------
// Precompute bias by relative offset: biasline[h][rel+2048], rel = k - q.
// ---------------------------------------------------------------------------
__global__ void bias_precompute(const float* __restrict__ bias_table,
                                float* __restrict__ biasline) {
  int idx = blockIdx.x * blockDim.x + threadIdx.x;
  if (idx >= Hc * 4096) return;
  int h = idx >> 12;
  int r = idx & 4095;
  int rel = r - 2048;   // k - q
  int n = -rel;         // q - k
  int ret = 0;
  if (n < 0) { ret = 16; n = -n; }
  int val;
  if (n < 8) {
    val = n;
  } else {
    float nf = (float)n;
    int vl = 8 + (int)(__logf(nf * 0.125f) * 0.3606737602222409f * 8.0f);
    val = vl < 15 ? vl : 15;
  }
  int bucket = ret + val;
  biasline[idx] = bias_table[(size_t)(bucket + 2048) * Hc + h];
}

// ---------------------------------------------------------------------------
// Fused flash attention, TDM double-buffered 64-key stages.
// grid = (S/64, H, B), block = 128 (4 waves); wave w owns q rows
// [64*bx + 16w, +16). Q' = [Q|PosQ] (16x128) pre-scaled by 1/sqrt(128).
// K'|PosK tile DMA'd by TDM with 68-dword row padding -> [64][132] LDS layout
// (bank-conflict-free B-fragment reads). V tile DMA'd unpadded [64][64].
// ---------------------------------------------------------------------------
__global__ __launch_bounds__(128) void tupe_attn(
    const float* __restrict__ tok,      // [B,S,3D]  (key | query | value)
    const float* __restrict__ poskq,    // [S,2D]    (pos_key | pos_query)
    const float* __restrict__ biasline, // [H,4096]
    float* __restrict__ out) {          // [B,S,D]
  __shared__ float Kc[2][64][132];   // [buf][key][feat 0..127 (tokK|posK)] pad 4
  __shared__ float Vs[2][64][64];
  __shared__ float Ps[4][16][68];    // per-wave P scratch (C -> A relayout)

  const int tid = threadIdx.x;
  const int wave = tid >> 5;
  const int lane = tid & 31;
  const int half = lane >> 4;
  const int l = lane & 15;
  const int h = blockIdx.y;
  const int b = blockIdx.z;
  const int qbase = blockIdx.x * 64 + wave * 16;
  const float inv = 0.08838834764831845f;   // 1/sqrt(2*DH)

  // ---- Load the 16x128 [Q|PosQ] A-operand, pre-scaled ----
  v2f aQ[32];
  {
    int qrow = qbase + l;
    const float* tq = tok + ((size_t)b * Sc + qrow) * (3 * Dc) + Dc + h * DHc;
    const float* pq = poskq + (size_t)qrow * (2 * Dc) + Dc + h * DHc;
    for (int t = 0; t < 16; ++t) {
      int c0 = t * 4 + 2 * half;
      aQ[t].x = tq[c0] * inv;
      aQ[t].y = tq[c0 + 1] * inv;
    }
    for (int t = 16; t < 32; ++t) {
      int c0 = (t - 16) * 4 + 2 * half;
      aQ[t].x = pq[c0] * inv;
      aQ[t].y = pq[c0 + 1] * inv;
    }
  }

  v8f O[4] = {};
  float m[8], lsum[8];
  for (int j = 0; j < 8; ++j) { m[j] = -1e30f; lsum[j] = 0.0f; }
  const float* bl = biasline + (size_t)h * 4096;

  constexpr int NT = Sc / 64;   // 32 stages of 64 keys

  // ---- prologue: DMA tile 0 into buffer 0 ----
  if (wave == 0) {
    const float* gK = tok + ((size_t)b * Sc) * (3 * Dc) + h * DHc;
    const float* gP = poskq + h * DHc;
    const float* gV = tok + ((size_t)b * Sc) * (3 * Dc) + 2 * Dc + h * DHc;
    unsigned kbase = (unsigned)(uintptr_t)&Kc[0][0][0];
    unsigned vbase = (unsigned)(uintptr_t)&Vs[0][0][0];
    tdm_load_2d(kbase,       gK, 64, 64, 3 * Dc, 1, 5, 67);  // pad 68 after 64
    tdm_load_2d(kbase + 256, gP, 64, 64, 2 * Dc, 1, 5, 67);  // interleave posK
    tdm_load_2d(vbase,       gV, 64, 64, 3 * Dc, 0, 0, 0);
    __builtin_amdgcn_s_wait_tensorcnt(0);
  }
  __syncthreads();

  for (int kt = 0; kt < NT; ++kt) {
    const int cur = kt & 1;
    const int k0 = kt * 64;

    // ---- issue next tile's DMA into the other buffer (overlaps compute) ----
    if (wave == 0 && kt + 1 < NT) {
      const int kn = (kt + 1) * 64;
      const float* gK = tok + ((size_t)b * Sc + kn) * (3 * Dc) + h * DHc;
      const float* gP = poskq + (size_t)kn * (2 * Dc) + h * DHc;
      const float* gV = tok + ((size_t)b * Sc + kn) * (3 * Dc) + 2 * Dc + h * DHc;
      unsigned kbase = (unsigned)(uintptr_t)&Kc[cur ^ 1][0][0];
      unsigned vbase = (unsigned)(uintptr_t)&Vs[cur ^ 1][0][0];
      tdm_load_2d(kbase,       gK, 64, 64, 3 * Dc, 1, 5, 67);
      tdm_load_2d(kbase + 256, gP, 64, 64, 2 * Dc, 1, 5, 67);
      tdm_load_2d(vbase,       gV, 64, 64, 3 * Dc, 0, 0, 0);
    }

    // ---- scores: 4 x [ S(16x16) = Q'(16x128) * K'^T ] -> 128 WMMAs ----
    v8f s[4];
    for (int t2 = 0; t2 < 4; ++t2) s[t2] = v8f{};
    for (int t2 = 0; t2 < 4; ++t2) {
      for (int t = 0; t < 32; ++t) {
        int c0 = t * 4 + 2 * half;
        v2f bv;
        bv.x = Kc[cur][t2 * 16 + l][c0];
        bv.y = Kc[cur][t2 * 16 + l][c0 + 1];
        s[t2] = wmma_f32(aQ[t], bv, s[t2]);
      }
    }

    // ---- bias + online softmax over 64 keys ----
    float sv[4][8];
    for (int t2 = 0; t2 < 4; ++t2)
      for (int j = 0; j < 8; ++j) {
        int qpos = qbase + j + 8 * half;
        int kpos = k0 + t2 * 16 + l;
        sv[t2][j] = s[t2][j] + bl[kpos - qpos + 2048];
      }
    float p[4][8], rescale[8];
    for (int j = 0; j < 8; ++j) {
      float mx = fmaxf(fmaxf(sv[0][j], sv[1][j]), fmaxf(sv[2][j], sv[3][j]));
      for (int off = 1; off < 16; off <<= 1)
        mx = fmaxf(mx, __shfl_xor(mx, off, 32));   // within 16-lane half
      float mnew = fmaxf(m[j], mx);
      float sc = __expf(m[j] - mnew);
      float rs = 0.0f;
      for (int t2 = 0; t2 < 4; ++t2) {
        float pv = __expf(sv[t2][j] - mnew);
        p[t2][j] = pv;
        rs += pv;
      }
      for (int off = 1; off < 16; off <<= 1)
        rs += __shfl_xor(rs, off, 32);
      lsum[j] = lsum[j] * sc + rs;
      m[j] = mnew;
      rescale[j] = sc;
    }
    for (int f = 0; f < 4; ++f)
      for (int j = 0; j < 8; ++j)
        O[f][j] *= rescale[j];

    // ---- relayout P (C layout) -> A fragments via LDS ----
    for (int t2 = 0; t2 < 4; ++t2)
      for (int j = 0; j < 8; ++j)
        Ps[wave][j + 8 * half][t2 * 16 + l] = p[t2][j];
    __syncthreads();
    v2f aP[16];
    for (int t = 0; t < 16; ++t) {
      aP[t].x = Ps[wave][l][t * 4 + 2 * half + 0];
      aP[t].y = Ps[wave][l][t * 4 + 2 * half + 1];
    }

    // ---- O(16x64) += P(16x64) * V(64x64) -> 64 WMMAs ----
    for (int t = 0; t < 16; ++t) {
      for (int f = 0; f < 4; ++f) {
        v2f bv;
        bv.x = Vs[cur][t * 4 + 2 * half + 0][f * 16 + l];
        bv.y = Vs[cur][t * 4 + 2 * half + 1][f * 16 + l];
        O[f] = wmma_f32(aP[t], bv, O[f]);
      }
    }

    // ---- drain next tile's DMA, then release both buffers ----
    if (wave == 0) __builtin_amdgcn_s_wait_tensorcnt(0);
    __syncthreads();
  }

  // ---- normalize and store ----
  for (int j = 0; j < 8; ++j) {
    float rl = 1.0f / lsum[j];
    int qpos = qbase + j + 8 * half;
    for (int f = 0; f < 4; ++f)
      out[((size_t)b * Sc + qpos) * Dc + h * DHc + f * 16 + l] = O[f][j] * rl;
  }
}

// ---------------------------------------------------------------------------
extern "C" void kernel_launch(void* const* d_in, const int* in_sizes, int n_in,
                              void* d_out, int out_size, void* d_ws, size_t ws_size,
                              hipStream_t stream) {
  const float* x          = (const float*)d_in[0];   // [B,S,D]
  const float* pos_embed  = (const float*)d_in[1];   // [S,D]
  const float* W_pos_kq   = (const float*)d_in[2];   // [D,2D]
  const float* W_tok_kqv  = (const float*)d_in[3];   // [D,3D]
  const float* bias_table = (const float*)d_in[4];   // [2*MAX_LEN,H]
  float* out = (float*)d_out;

  float* ws     = (float*)d_ws;
  float* tok    = ws;                                 // B*S*3D floats (48 MB)
  float* poskq  = tok + (size_t)Bc * Sc * 3 * Dc;     // S*2D floats (16 MB)
  float* biasln = poskq + (size_t)Sc * 2 * Dc;        // H*4096 floats (256 KB)

  wmma_gemm_f32<<<dim3((3 * Dc) / 64, (Bc * Sc) / 64), dim3(128), 0, stream>>>(
      x, W_tok_kqv, tok, Bc * Sc, 3 * Dc, Dc);
  wmma_gemm_f32<<<dim3((2 * Dc) / 64, Sc / 64), dim3(128), 0, stream>>>(
      pos_embed, W_pos_kq, poskq, Sc, 2 * Dc, Dc);
  bias_precompute<<<(Hc * 4096) / 256, 256, 0, stream>>>(bias_table, biasln);
  tupe_attn<<<dim3(Sc / 64, Hc, Bc), dim3(128), 0, stream>>>(tok, poskq, biasln, out);
}